// matrix_times_41583873359887
// MI455X (gfx1250) — compile-verified
//
#include <hip/hip_runtime.h>

typedef __attribute__((ext_vector_type(2))) float    v2f;
typedef __attribute__((ext_vector_type(4))) float    v4f;
typedef __attribute__((ext_vector_type(8))) float    v8f;
typedef __attribute__((ext_vector_type(4))) unsigned u32x4;
typedef __attribute__((ext_vector_type(8))) int      i32x8;
typedef __attribute__((ext_vector_type(4))) int      i32x4;

#define DIM 4096
#define BM 128
#define BN 128
#define BK 32
#define NSTAGES (DIM / BK)
#define LDA 36    // A LDS pitch (floats): 32 data + 4 pad (TDM pad_interval=32dw, pad_amount=4dw)
#define LDBP 288  // B paired pitch (floats): 288 % 64 == 32 -> half-waves hit disjoint bank halves

// byte offset of a generic LDS pointer within the workgroup's LDS segment
__device__ __forceinline__ unsigned lds_byte_off(const void* p) {
    return (unsigned)(unsigned long long)(__attribute__((address_space(3))) const void*)p;
}

// ---- Tensor Data Mover: DMA one 128x32 f32 tile (row stride DIM) into padded LDS ----
__device__ __forceinline__ void tdm_load_A_tile(unsigned lds_off, const float* gtile) {
    const unsigned long long ga = (unsigned long long)gtile;

    u32x4 g0;
    g0[0] = 1u;                                              // count=1, user mode, no gather
    g0[1] = lds_off;                                         // lds_addr (bytes)
    g0[2] = (unsigned)ga;                                    // global_addr[31:0]
    g0[3] = (unsigned)((ga >> 32) & 0x1FFFFFFu) | (2u << 30); // global_addr[56:32] | type=2

    i32x8 g1;
    g1[0] = (2 << 16)      // data_size = 4 bytes
          | (1 << 20)      // pad_enable (LDS dest padding)
          | (4 << 22)      // pad_interval code 4 -> every 32 DWORDs
          | (3 << 25);     // pad_amount  code 3 -> 4 DWORDs
    g1[1] = (int)((DIM & 0xFFFF) << 16);   // tensor_dim0[15:0]=4096 in bits[31:16]; barrier_addr=0
    g1[2] = (int)((DIM & 0xFFFF) << 16);   // tensor_dim0[31:16]=0 | tensor_dim1[15:0]=4096
    g1[3] = (int)(BK << 16);               // tensor_dim1[31:16]=0 | tile_dim0=32
    g1[4] = BM;                            // tile_dim1=128 | tile_dim2=0
    g1[5] = DIM;                           // tensor_dim0_stride[31:0]=4096
    g1[6] = 0;                             // stride[47:32]=0 | tensor_dim1_stride[15:0]=0
    g1[7] = 0;

    i32x4 g2 = {0, 0, 0, 0};               // 2D tile: dims 2+ unused
    i32x4 g3 = {0, 0, 0, 0};

#if __has_builtin(__builtin_amdgcn_tensor_load_to_lds)
  #if defined(__clang_major__) && __clang_major__ >= 23
    i32x8 g4 = {0, 0, 0, 0, 0, 0, 0, 0};
    __builtin_amdgcn_tensor_load_to_lds(g0, g1, g2, g3, g4, 0);
  #else
    __builtin_amdgcn_tensor_load_to_lds(g0, g1, g2, g3, 0);
  #endif
#endif
}

__device__ __forceinline__ void wait_tensor0() {
#if __has_builtin(__builtin_amdgcn_s_wait_tensorcnt)
    __builtin_amdgcn_s_wait_tensorcnt(0);
#else
    asm volatile("s_wait_tensorcnt 0x0" ::: "memory");
#endif
}

__global__ __launch_bounds__(256, 2)
void wmma_gemm_f32_tdm(const float* __restrict__ A,   // jacobian, M x K
                       const float* __restrict__ B,   // eye,      K x N
                       float* __restrict__ C) {       // M x N
    __shared__ float As[2][BM * LDA];        // 2 x 18432 B, TDM double buffer
    __shared__ float Bs[(BK / 2) * LDBP];    // 18432 B, K-paired layout

    const int tid    = threadIdx.x;
    const int lane   = tid & 31;
    const int laneLo = lane & 15;
    const int laneHi = lane >> 4;            // half-wave (K split per ISA layout)
    const int wave   = tid >> 5;
    const int waveM  = wave >> 2;            // 0..1
    const int waveN  = wave & 3;             // 0..3

    const int blockRow = blockIdx.y * BM;
    const int blockCol = blockIdx.x * BN;

    // B staging: 2 passes per thread, each pairs rows (2kp, 2kp+1)
    const int bkp  = tid >> 5;               // + p*8  -> pair index 0..15
    const int bcol = (tid & 31) * 4;

    const float* Atile = A + (size_t)blockRow * DIM;   // advance by BK each stage
    const float* Bbase = B + blockCol + bcol;

    v8f acc[4][2] = {};
    v4f b0[2], b1[2];

    // ---------------- prologue: stage 0 ----------------
    if (wave == 0)
        tdm_load_A_tile(lds_byte_off(&As[0][0]), Atile);

    #pragma unroll
    for (int p = 0; p < 2; ++p) {
        const int kp = bkp + p * 8;
        b0[p] = *(const v4f*)(Bbase + (size_t)(2 * kp) * DIM);
        b1[p] = *(const v4f*)(Bbase + (size_t)(2 * kp + 1) * DIM);
    }
    #pragma unroll
    for (int p = 0; p < 2; ++p) {
        const int kp = bkp + p * 8;
        #pragma unroll
        for (int i = 0; i < 4; ++i) {
            v2f pr; pr[0] = b0[p][i]; pr[1] = b1[p][i];
            *(v2f*)&Bs[kp * LDBP + 2 * (bcol + i)] = pr;   // 8B-aligned b64 store
        }
    }
    if (wave == 0) wait_tensor0();
    __syncthreads();

    // ---------------- pipelined main loop ----------------
    int buf = 0;
    for (int s = 0; s < NSTAGES; ++s) {
        const bool hasNext = (s + 1) < NSTAGES;
        const int  k0n     = (s + 1) * BK;

        if (hasNext) {
            if (wave == 0)
                tdm_load_A_tile(lds_byte_off(&As[buf ^ 1][0]), Atile + k0n);
            #pragma unroll
            for (int p = 0; p < 2; ++p) {
                const int kp = bkp + p * 8;
                b0[p] = *(const v4f*)(Bbase + (size_t)(k0n + 2 * kp) * DIM);
                b1[p] = *(const v4f*)(Bbase + (size_t)(k0n + 2 * kp + 1) * DIM);
            }
        }

        // ---- compute current stage: 8 K-steps x 8 tiles = 64 WMMAs ----
        const float* Asb = As[buf];
        #pragma unroll
        for (int kk = 0; kk < BK; kk += 4) {
            const int k  = kk + 2 * laneHi;
            const int kp = (kk >> 1) + laneHi;

            v2f afrag[4];
            #pragma unroll
            for (int tm = 0; tm < 4; ++tm) {
                const int m = waveM * 64 + tm * 16 + laneLo;
                afrag[tm] = *(const v2f*)&Asb[m * LDA + k];        // V0=K=k, V1=K=k+1
            }
            v2f bfrag[2];
            #pragma unroll
            for (int tn = 0; tn < 2; ++tn) {
                const int n = waveN * 32 + tn * 16 + laneLo;
                bfrag[tn] = *(const v2f*)&Bs[kp * LDBP + 2 * n];   // pre-paired b64
            }
            #pragma unroll
            for (int tm = 0; tm < 4; ++tm) {
                #pragma unroll
                for (int tn = 0; tn < 2; ++tn) {
                    acc[tm][tn] = __builtin_amdgcn_wmma_f32_16x16x4_f32(
                        false, afrag[tm], false, bfrag[tn],
                        (short)0, acc[tm][tn], false, false);
                }
            }
        }
        __syncthreads();                 // all waves done reading Bs / As[buf]

        if (hasNext) {
            #pragma unroll
            for (int p = 0; p < 2; ++p) {
                const int kp = bkp + p * 8;
                #pragma unroll
                for (int i = 0; i < 4; ++i) {
                    v2f pr; pr[0] = b0[p][i]; pr[1] = b1[p][i];
                    *(v2f*)&Bs[kp * LDBP + 2 * (bcol + i)] = pr;
                }
            }
            if (wave == 0) wait_tensor0();   // TDM tile for next stage has landed
            __syncthreads();                 // publish B stores + TDM data to all waves
            buf ^= 1;
        }
    }

    // ---------------- epilogue: C/D layout -> VGPR r = rows r / r+8 per half-wave ----------------
    #pragma unroll
    for (int tm = 0; tm < 4; ++tm) {
        #pragma unroll
        for (int tn = 0; tn < 2; ++tn) {
            const int row0 = blockRow + waveM * 64 + tm * 16 + laneHi * 8;
            const int col  = blockCol + waveN * 32 + tn * 16 + laneLo;
            #pragma unroll
            for (int r = 0; r < 8; ++r) {
                C[(size_t)(row0 + r) * DIM + col] = acc[tm][tn][r];
            }
        }
    }
}

extern "C" void kernel_launch(void* const* d_in, const int* in_sizes, int n_in,
                              void* d_out, int out_size, void* d_ws, size_t ws_size,
                              hipStream_t stream) {
    // setup_inputs order: {"eye", "jacobian"}; result = jacobian @ eye
    const float* eye = (const float*)d_in[0];
    const float* jac = (const float*)d_in[1];
    float* out = (float*)d_out;

    dim3 grid(DIM / BN, DIM / BM);   // 32 x 32 blocks
    dim3 block(256, 1, 1);           // 8 wave32 waves per block
    wmma_gemm_f32_tdm<<<grid, block, 0, stream>>>(jac, eye, out);
}